// PatternANeoXRoPE_81776177315982
// MI455X (gfx1250) — compile-verified
//
#include <hip/hip_runtime.h>

typedef float v2f __attribute__((ext_vector_type(2)));
typedef float v8f __attribute__((ext_vector_type(8)));

#define BB 2
#define SS 2048
#define HH 16
#define DD 128
#define QB 128            // queries per workgroup (8 waves x 16)
#define KSTRIDE 132       // LDS stride (floats) for K tile rows  (bank-friendly)
#define VSTRIDE 18        // LDS stride (floats) for transposed V (8B-aligned b64)
#define PSTRIDE 18        // LDS stride (floats) for P scratch

__global__ __launch_bounds__(256)
void fa_rope_fwd(const float* __restrict__ q, const float* __restrict__ k,
                 const float* __restrict__ v, const float* __restrict__ cosp,
                 const float* __restrict__ sinp, float* __restrict__ out)
{
    __shared__ float kbuf[16 * KSTRIDE];          // roped K tile, row-major [key][d]
    __shared__ float vbufT[DD * VSTRIDE];         // V tile transposed   [d][key]
    __shared__ float pbuf[8 * 16 * PSTRIDE];      // per-wave P scratch  [m][k]

    const int tid  = threadIdx.x;
    const int wave = tid >> 5;
    const int lane = tid & 31;
    const int lmod = lane & 15;       // M (A/C row sel) or N (B/C col sel)
    const int hi   = lane >> 4;       // half-wave
    const int off  = hi * 2;          // K-offset inside each 4-group

    const int qblk = blockIdx.x & 15;             // 16 q-blocks of 128
    const int bh   = blockIdx.x >> 4;             // b*H + h  (0..31)
    const int b    = bh >> 4;
    const int h    = bh & 15;

    const int qbase = qblk * QB + wave * 16;      // first query row of this wave
    const int myrow = qbase + lmod;               // this lane's query position s

    // ---- Q row -> registers in WMMA A layout, NeoX RoPE in-register ----
    // lane needs dims d = 4*st + off + {0,1}; partner d+-64 is in same class.
    float qa[64];
    {
        const float* qrow = q + (((size_t)b * SS + myrow) * HH + h) * DD;
        const float* crow = cosp + (size_t)myrow * DD;
        const float* srow = sinp + (size_t)myrow * DD;
        float raw[64];
#pragma unroll
        for (int st = 0; st < 32; ++st) {
            v2f t = *(const v2f*)(qrow + 4 * st + off);
            raw[2 * st] = t.x; raw[2 * st + 1] = t.y;
        }
        const float scale = 0.08838834764831845f;  // 1/sqrt(128)
#pragma unroll
        for (int st = 0; st < 16; ++st) {
            const int dlo = 4 * st + off, dhi = dlo + 64;
            v2f cl = *(const v2f*)(crow + dlo);
            v2f sl = *(const v2f*)(srow + dlo);
            v2f ch = *(const v2f*)(crow + dhi);
            v2f sh = *(const v2f*)(srow + dhi);
            const int lo = 2 * st, hp = lo + 32;
            qa[lo]     = (raw[lo]     * cl.x - raw[hp]     * sl.x) * scale;
            qa[lo + 1] = (raw[lo + 1] * cl.y - raw[hp + 1] * sl.y) * scale;
            qa[hp]     = (raw[hp]     * ch.x + raw[lo]     * sh.x) * scale;
            qa[hp + 1] = (raw[hp + 1] * ch.y + raw[lo + 1] * sh.y) * scale;
        }
    }

    // ---- accumulators (C/D layout: lane col = lmod, rows r+8*hi) ----
    v8f o[8];
#pragma unroll
    for (int j = 0; j < 8; ++j)
#pragma unroll
        for (int r = 0; r < 8; ++r) o[j][r] = 0.0f;
    float mrow[8], lrow[8];
#pragma unroll
    for (int r = 0; r < 8; ++r) { mrow[r] = -3.0e38f; lrow[r] = 0.0f; }

    const int ntiles  = qblk * 8 + 8;     // staged by the whole block
    const int ktmax_w = qblk * 8 + wave;  // last tile this wave consumes (diag)
    const size_t kvhead = ((size_t)b * SS * HH + h) * DD;

    for (int kt = 0; kt < ntiles; ++kt) {
        const int kbase = kt * 16;
        __syncthreads();
        // ---- cooperative stage: K tile (roped) + V tile (transposed) ----
        {
            const int row = tid >> 4;
            const int c4  = (tid & 15) * 4;
            const int sk  = kbase + row;
            const float* krow = k + kvhead + (size_t)sk * HH * DD;
            const float* vrow = v + kvhead + (size_t)sk * HH * DD;
            const float* crow = cosp + (size_t)sk * DD;
            const float* srow = sinp + (size_t)sk * DD;
            float4 ka = *(const float4*)(krow + c4);
            float4 kb = *(const float4*)(krow + c4 + 64);
            float4 ca = *(const float4*)(crow + c4);
            float4 cb = *(const float4*)(crow + c4 + 64);
            float4 sa = *(const float4*)(srow + c4);
            float4 sb = *(const float4*)(srow + c4 + 64);
            float4 ra, rb;
            ra.x = ka.x * ca.x - kb.x * sa.x;  ra.y = ka.y * ca.y - kb.y * sa.y;
            ra.z = ka.z * ca.z - kb.z * sa.z;  ra.w = ka.w * ca.w - kb.w * sa.w;
            rb.x = kb.x * cb.x + ka.x * sb.x;  rb.y = kb.y * cb.y + ka.y * sb.y;
            rb.z = kb.z * cb.z + ka.z * sb.z;  rb.w = kb.w * cb.w + ka.w * sb.w;
            *(float4*)(kbuf + row * KSTRIDE + c4)      = ra;
            *(float4*)(kbuf + row * KSTRIDE + c4 + 64) = rb;
            float4 va = *(const float4*)(vrow + c4);
            float4 vb = *(const float4*)(vrow + c4 + 64);
            vbufT[(c4 + 0) * VSTRIDE + row] = va.x;
            vbufT[(c4 + 1) * VSTRIDE + row] = va.y;
            vbufT[(c4 + 2) * VSTRIDE + row] = va.z;
            vbufT[(c4 + 3) * VSTRIDE + row] = va.w;
            vbufT[(c4 + 64) * VSTRIDE + row] = vb.x;
            vbufT[(c4 + 65) * VSTRIDE + row] = vb.y;
            vbufT[(c4 + 66) * VSTRIDE + row] = vb.z;
            vbufT[(c4 + 67) * VSTRIDE + row] = vb.w;
        }
        __syncthreads();

        if (kt <= ktmax_w) {                  // wave-uniform; EXEC stays full
            // ---- S = Q * K^T : 32 chained f32 WMMAs over the D dimension ----
            v8f sc;
#pragma unroll
            for (int r = 0; r < 8; ++r) sc[r] = 0.0f;
#pragma unroll
            for (int st = 0; st < 32; ++st) {
                v2f a;  a.x = qa[2 * st]; a.y = qa[2 * st + 1];
                v2f bb = *(const v2f*)(kbuf + lmod * KSTRIDE + 4 * st + off);
                sc = __builtin_amdgcn_wmma_f32_16x16x4_f32(
                        false, a, false, bb, (short)0, sc, false, false);
            }
            // ---- causal mask on the diagonal tile (elementwise select) ----
            if (kt == ktmax_w) {
#pragma unroll
                for (int r = 0; r < 8; ++r) {
                    const int m = r + 8 * hi;
                    sc[r] = (lmod > m) ? -3.0e38f : sc[r];
                }
            }
            // ---- online softmax: row reductions across 16-lane groups ----
            float alpha[8];
            v8f p;
#pragma unroll
            for (int r = 0; r < 8; ++r) {
                float t = sc[r];
                t = fmaxf(t, __shfl_xor(t, 1));
                t = fmaxf(t, __shfl_xor(t, 2));
                t = fmaxf(t, __shfl_xor(t, 4));
                t = fmaxf(t, __shfl_xor(t, 8));
                const float mn = fmaxf(mrow[r], t);
                alpha[r] = __expf(mrow[r] - mn);
                mrow[r]  = mn;
                const float pe = __expf(sc[r] - mn);
                p[r] = pe;
                float su = pe;
                su += __shfl_xor(su, 1);
                su += __shfl_xor(su, 2);
                su += __shfl_xor(su, 4);
                su += __shfl_xor(su, 8);
                lrow[r] = lrow[r] * alpha[r] + su;
            }
#pragma unroll
            for (int j = 0; j < 8; ++j)
#pragma unroll
                for (int r = 0; r < 8; ++r) o[j][r] *= alpha[r];

            // ---- C-layout -> A-layout for P via per-wave LDS scratch ----
            float* pb = pbuf + wave * 16 * PSTRIDE;
#pragma unroll
            for (int r = 0; r < 8; ++r)
                pb[(r + 8 * hi) * PSTRIDE + lmod] = p[r];
            v2f ap[4];
#pragma unroll
            for (int k0 = 0; k0 < 4; ++k0)
                ap[k0] = *(const v2f*)(pb + lmod * PSTRIDE + 4 * k0 + off);

            // ---- O += P * V : 32 f32 WMMAs (8 dim-tiles x 4 K-steps) ----
#pragma unroll
            for (int j = 0; j < 8; ++j) {
#pragma unroll
                for (int k0 = 0; k0 < 4; ++k0) {
                    v2f bb = *(const v2f*)(vbufT + (16 * j + lmod) * VSTRIDE + 4 * k0 + off);
                    o[j] = __builtin_amdgcn_wmma_f32_16x16x4_f32(
                               false, ap[k0], false, bb, (short)0, o[j], false, false);
                }
            }
        }
    }

    // ---- epilogue: O / l, write (B,H,S,D) ----
    float inv_l[8];
#pragma unroll
    for (int r = 0; r < 8; ++r) inv_l[r] = 1.0f / lrow[r];
    float* obase = out + (size_t)bh * SS * DD;
#pragma unroll
    for (int j = 0; j < 8; ++j) {
#pragma unroll
        for (int r = 0; r < 8; ++r) {
            const int m = r + 8 * hi;
            obase[(size_t)(qbase + m) * DD + 16 * j + lmod] = o[j][r] * inv_l[r];
        }
    }
}

extern "C" void kernel_launch(void* const* d_in, const int* in_sizes, int n_in,
                              void* d_out, int out_size, void* d_ws, size_t ws_size,
                              hipStream_t stream) {
    (void)in_sizes; (void)n_in; (void)out_size; (void)d_ws; (void)ws_size;
    const float* q = (const float*)d_in[0];
    const float* k = (const float*)d_in[1];
    const float* v = (const float*)d_in[2];
    const float* c = (const float*)d_in[3];
    const float* s = (const float*)d_in[4];
    float* o = (float*)d_out;
    dim3 grid(BB * HH * (SS / QB));   // 512 workgroups
    dim3 block(256);                  // 8 wave32 waves
    hipLaunchKernelGGL(fa_rope_fwd, grid, block, 0, stream, q, k, v, c, s, o);
}